// SigFormer_48473000902849
// MI455X (gfx1250) — compile-verified
//
#include <hip/hip_runtime.h>

typedef __attribute__((ext_vector_type(2))) float v2f;
typedef __attribute__((ext_vector_type(8))) float v8f;

static constexpr int Bn = 512, Nn = 1024, Dn = 8;

// ---------------------------------------------------------------------------
// Prep: fold W_sig @ (W_v @ (W_o @ W_out)) into u[72] plus scalars Kc, K0.
// ws[0..71] = u (u1 = ws[0..7], U[a][c] = ws[8 + a*8 + c]), ws[72]=Kc, ws[73]=K0
// ---------------------------------------------------------------------------
__global__ __launch_bounds__(64) void sigformer_prep(
    const float* __restrict__ W_sig, const float* __restrict__ b_sig,
    const float* __restrict__ W_v,   const float* __restrict__ b_v,
    const float* __restrict__ W_o,   const float* __restrict__ b_o,
    const float* __restrict__ W_out, const float* __restrict__ b_out,
    float* __restrict__ ws)
{
  __shared__ float q[64], weff[64];
  const int i = threadIdx.x;

  float acc = 0.f;                       // q = W_o @ W_out
  for (int j = 0; j < 64; ++j) acc += W_o[i*64 + j] * W_out[j];
  q[i] = acc;
  __syncthreads();

  float acc2 = 0.f;                      // weff = W_v @ q
  for (int j = 0; j < 64; ++j) acc2 += W_v[i*64 + j] * q[j];
  weff[i] = acc2;
  __syncthreads();

  for (int s = i; s < 72; s += 64) {     // u = W_sig @ weff
    float a = 0.f;
    for (int k = 0; k < 64; ++k) a += W_sig[s*64 + k] * weff[k];
    ws[s] = a;
  }
  if (i == 0) {
    float K0 = b_out[0];
    for (int k = 0; k < 64; ++k) K0 += b_v[k]*q[k] + b_o[k]*W_out[k];
    float Kc = K0;
    for (int k = 0; k < 64; ++k) Kc += b_sig[k]*weff[k];
    ws[72] = Kc; ws[73] = K0;
  }
}

// ---------------------------------------------------------------------------
// Main: one 256-thread block (8 wave32) per batch row; wave w owns timesteps
// [w*128, w*128+128) as 8 unrolled tiles of 16. Per tile:
//   Y = U_pad(16x8) @ inc^T(8x16)   (2x v_wmma_f32_16x16x4_f32)
//   T = G(16x8)     @ Y(8x16)       (2x v_wmma_f32_16x16x4_f32)
//   s_j = T[j][j];  delta_part = u1.G + local_prefix(s) + Kc
// s_j depends only on j (C_j = f[j]-f[0]), so chunks combine with a plain
// 8-entry LDS scan — no sequential cross-chunk terms.
// ---------------------------------------------------------------------------
__global__ __launch_bounds__(256) void sigformer_main(
    const float* __restrict__ feat, const float* __restrict__ ws,
    float* __restrict__ out)
{
  __shared__ float ctot[8];

  const int lane = threadIdx.x & 31;
  const int wave = threadIdx.x >> 5;
  const int b    = blockIdx.x;
  const int r = lane & 15;     // matrix row (A) / column (B) owned by this lane
  const int h = lane >> 4;     // lane half selects K pair {0,1} vs {2,3}
  const int d = h * 2;
  const int tc = wave * 128;   // chunk start

  const float Kc = ws[72];
  const float K0 = ws[73];

  // U as A-operand (rows 8..15 zero): ISA f32 A 16x4 layout,
  // VGPR0 = K{0|2}, VGPR1 = K{1|3} by lane half.
  v2f uA_lo = {0.f, 0.f}, uA_hi = {0.f, 0.f};
  if (r < 8) {
    const float* up = ws + 8 + r*8;
    uA_lo.x = up[d];     uA_lo.y = up[d+1];
    uA_hi.x = up[d+4];   uA_hi.y = up[d+5];
  }
  const float u1a = ws[d], u1b = ws[d+1], u1c = ws[d+4], u1e = ws[d+5];

  const float* frow = feat + (size_t)b * (Nn * Dn);
  const float f0a = frow[d], f0b = frow[d+1], f0c = frow[d+4], f0e = frow[d+5];

  const v8f vzero = {0.f,0.f,0.f,0.f,0.f,0.f,0.f,0.f};

  float Sloc = 0.f;        // prefix within this wave's chunk
  float dpart[8];          // per-tile partial delta (valid in lanes 0..15)

  #pragma unroll
  for (int it = 0; it < 8; ++it) {
    const int t0 = tc + it * 16;
    const int t  = t0 + r;
    const float* p = frow + t * Dn;
    const int tn = (t + 1 < Nn) ? (t + 1) : t;      // clamp -> inc = 0 at t = N-1
    const float* pn = frow + tn * Dn;
    __builtin_prefetch((const void*)(p + 16*Dn), 0, 0);

    const float fa = p[d],  fb = p[d+1],  fc = p[d+4],  fe = p[d+5];
    const float na = pn[d], nb = pn[d+1], nc = pn[d+4], ne = pn[d+5];

    // inc^T as B-operand (B[k][n]: VGPR0 = k{0|2}, VGPR1 = k{1|3} by lane half)
    v2f incB_lo = { na - fa, nb - fb };
    v2f incB_hi = { nc - fc, ne - fe };
    // G = f[t]-f[0] as A-operand
    v2f gA_lo = { fa - f0a, fb - f0b };
    v2f gA_hi = { fc - f0c, fe - f0e };

    // Y = U_pad @ inc^T  (16x16, rows 0..7 valid)
    v8f Y = __builtin_amdgcn_wmma_f32_16x16x4_f32(false, uA_lo, false, incB_lo,
                                                  (short)0, vzero, false, false);
    Y     = __builtin_amdgcn_wmma_f32_16x16x4_f32(false, uA_hi, false, incB_hi,
                                                  (short)0, Y, false, false);

    // Re-layout Y rows (D layout: row k at VGPR k, lanes 0..15) into B-operand:
    // lane half 1 needs rows {2,3,6,7} from lanes l-16 -> shfl_xor 16.
    const float y2 = __shfl_xor(Y[2], 16);
    const float y3 = __shfl_xor(Y[3], 16);
    const float y6 = __shfl_xor(Y[6], 16);
    const float y7 = __shfl_xor(Y[7], 16);
    v2f yB_lo = { h ? y2 : Y[0], h ? y3 : Y[1] };
    v2f yB_hi = { h ? y6 : Y[4], h ? y7 : Y[5] };

    // T = G @ Y : T[j][n] = G_j^T U inc_n ; we need the diagonal
    v8f T = __builtin_amdgcn_wmma_f32_16x16x4_f32(false, gA_lo, false, yB_lo,
                                                  (short)0, vzero, false, false);
    T     = __builtin_amdgcn_wmma_f32_16x16x4_f32(false, gA_hi, false, yB_hi,
                                                  (short)0, T, false, false);

    // Diagonal T[j][j]: j<8 -> lane j, VGPR j ; j>=8 -> lane 16+j, VGPR j-8
    const int idx = ((lane < 8) ? lane : (lane - 24)) & 7;
    float dv = T[0];
    dv = (idx == 1) ? T[1] : dv;
    dv = (idx == 2) ? T[2] : dv;
    dv = (idx == 3) ? T[3] : dv;
    dv = (idx == 4) ? T[4] : dv;
    dv = (idx == 5) ? T[5] : dv;
    dv = (idx == 6) ? T[6] : dv;
    dv = (idx == 7) ? T[7] : dv;

    // Gather s_j into lane j (j = 0..15), zero elsewhere
    const int src = (r < 8) ? r : (16 + r);
    float sj = __shfl(dv, src);
    sj = (lane < 16) ? sj : 0.f;

    // Wave inclusive scan -> exclusive prefix + tile total
    float x = sj;
    #pragma unroll
    for (int off = 1; off < 32; off <<= 1) {
      const float nbr = __shfl_up(x, (unsigned)off);
      x += (lane >= off) ? nbr : 0.f;
    }
    const float excl  = x - sj;
    const float total = __shfl(x, 31);   // lanes >=16 contributed 0

    // u1 . G  (split across lane pair l, l^16)
    const float part = u1a*gA_lo.x + u1b*gA_lo.y + u1c*gA_hi.x + u1e*gA_hi.y;
    const float dot  = part + __shfl_xor(part, 16);

    dpart[it] = dot + Sloc + excl + Kc;
    Sloc += total;
  }

  // Combine chunk totals across the 8 waves of this block.
  if (lane == 0) ctot[wave] = Sloc;
  __syncthreads();
  float base = 0.f;
  #pragma unroll
  for (int w2 = 0; w2 < 8; ++w2) base += (w2 < wave) ? ctot[w2] : 0.f;

  // Store pass (register-resident partials).
  #pragma unroll
  for (int it = 0; it < 8; ++it) {
    const int t = tc + it * 16 + r;
    float delta = dpart[it] + base;
    delta = (t == 0) ? K0 : delta;
    if (lane < 16) out[(size_t)b * Nn + t] = delta;
  }
}

// ---------------------------------------------------------------------------
extern "C" void kernel_launch(void* const* d_in, const int* in_sizes, int n_in,
                              void* d_out, int out_size, void* d_ws, size_t ws_size,
                              hipStream_t stream) {
  const float* feat  = (const float*)d_in[0];
  const float* W_sig = (const float*)d_in[1];
  const float* b_sig = (const float*)d_in[2];
  // d_in[3..8] (W_feat, b_feat, W_q, b_q, W_k, b_k) are dead code: softmax over
  // a single key makes attn == v, so deltas == deltas_pre exactly.
  const float* W_v   = (const float*)d_in[9];
  const float* b_v   = (const float*)d_in[10];
  const float* W_o   = (const float*)d_in[11];
  const float* b_o   = (const float*)d_in[12];
  const float* W_out = (const float*)d_in[13];
  const float* b_out = (const float*)d_in[14];
  float* ws  = (float*)d_ws;
  float* out = (float*)d_out;

  sigformer_prep<<<1, 64, 0, stream>>>(W_sig, b_sig, W_v, b_v, W_o, b_o,
                                       W_out, b_out, ws);
  sigformer_main<<<Bn, 256, 0, stream>>>(feat, ws, out);
}